// NodeLayer_81123342287184
// MI455X (gfx1250) — compile-verified
//
#include <hip/hip_runtime.h>
#include <hip/hip_bf16.h>

#define VN 512
#define CN 64
#define HN 16
#define DN 4
#define EN 3072
#define EFN 4096

typedef __attribute__((ext_vector_type(2))) float v2f;
typedef __attribute__((ext_vector_type(8))) float v8f;

// ---------------- wave32 reductions ----------------
__device__ inline float wave_sum(float v) {
#pragma unroll
  for (int o = 16; o > 0; o >>= 1) v += __shfl_xor(v, o, 32);
  return v;
}
__device__ inline float wave_max(float v) {
#pragma unroll
  for (int o = 16; o > 0; o >>= 1) v = fmaxf(v, __shfl_xor(v, o, 32));
  return v;
}

// ---------------- generic row-major WMMA GEMM: D = A[MxK] @ B[KxN] ----------------
// grid = (N/16, M/64), block = 128 (4 waves, one 16x16 tile each). K % 4 == 0.
__global__ __launch_bounds__(128) void wmma_gemm_rm(
    const float* __restrict__ A, const float* __restrict__ B,
    float* __restrict__ D, int M, int N, int K) {
  const int wave = threadIdx.x >> 5;
  const int l    = threadIdx.x & 31;
  const int n0   = blockIdx.x * 16;
  const int m0   = (blockIdx.y * 4 + wave) * 16;
  const int mrow = l & 15;           // A row / D col lane index
  const int kp   = (l >> 4) * 2;     // K-slot pair select (ISA 16x4 f32 A layout)
  v8f acc = {};
  const float* Ar = A + (size_t)(m0 + mrow) * K;
  for (int k0 = 0; k0 < K; k0 += 4) {
    v2f a = *(const v2f*)(Ar + k0 + kp);   // contiguous k,k+1 -> one b64 load
    v2f b;
    b.x = B[(size_t)(k0 + kp) * N + n0 + mrow];
    b.y = B[(size_t)(k0 + kp + 1) * N + n0 + mrow];
    acc = __builtin_amdgcn_wmma_f32_16x16x4_f32(false, a, false, b, (short)0,
                                                acc, false, false);
  }
  const int mh = (l >> 4) * 8;
#pragma unroll
  for (int v = 0; v < 8; ++v)
    D[(size_t)(m0 + mh + v) * N + n0 + mrow] = acc[v];
}

// ---------------- channel-batched pathway GEMM ----------------
// featOut[x,z,c] = sum_y featIn[x,y,c] * adj[y,z]
// grid = (V/16, V/16, C/8), block = 256 (8 waves share the adj z-panel in LDS).
// The adj panel is staged memory->LDS with GLOBAL_LOAD_ASYNC_TO_LDS_B128
// (CDNA5 async path, ASYNCcnt), then consumed by the f32 WMMA k-loop.
__global__ __launch_bounds__(256) void pathway_gemm(
    const float* __restrict__ featIn, const float* __restrict__ adj,
    float* __restrict__ featOut) {
  __shared__ __align__(16) float Bs[VN * 16];   // adj[0:V, z0:z0+16]  (32 KB)
  const int x0 = blockIdx.x * 16;
  const int z0 = blockIdx.y * 16;
  const int c  = blockIdx.z * 8 + (threadIdx.x >> 5);

  // Async stage: 2048 x 16B transfers, 8 per thread. GVS mode:
  //   mem  = saddr(adj + z0) + voff ;  lds = Bs + ldsoff
  {
    const float* gbase = adj + z0;                       // uniform -> SGPR pair
    unsigned bsbase = (unsigned)(uintptr_t)(&Bs[0]);     // flat addr low 32 = LDS offset
    for (int idx = threadIdx.x; idx < VN * 4; idx += 256) {
      int kk = idx >> 2, nn4 = (idx & 3) << 2;
      unsigned voff   = (unsigned)((kk * VN + nn4) * 4);
      unsigned ldsoff = bsbase + (unsigned)((kk * 16 + nn4) * 4);
      asm volatile("global_load_async_to_lds_b128 %0, %1, %2"
                   :: "v"(ldsoff), "v"(voff), "s"(gbase) : "memory");
    }
    asm volatile("s_wait_asynccnt 0x0" ::: "memory");
  }
  __syncthreads();

  const int l    = threadIdx.x & 31;
  const int mrow = l & 15;
  const int kp   = (l >> 4) * 2;
  v8f acc = {};
  const float* Ar = featIn + (size_t)(x0 + mrow) * VN * CN + c;
  for (int k0 = 0; k0 < VN; k0 += 4) {
    v2f a, b;
    a.x = Ar[(size_t)(k0 + kp) * CN];
    a.y = Ar[(size_t)(k0 + kp + 1) * CN];
    b.x = Bs[(k0 + kp) * 16 + mrow];
    b.y = Bs[(k0 + kp + 1) * 16 + mrow];
    acc = __builtin_amdgcn_wmma_f32_16x16x4_f32(false, a, false, b, (short)0,
                                                acc, false, false);
  }
  const int mh = (l >> 4) * 8;
#pragma unroll
  for (int v = 0; v < 8; ++v)
    featOut[((size_t)(x0 + mh + v) * VN + z0 + mrow) * CN + c] = acc[v];
}

// ---------------- elementwise / scatter kernels ----------------
__global__ void scatter_edges(const int* __restrict__ eidx,
                              const float* __restrict__ dual_attr,
                              float* __restrict__ adj0, float* __restrict__ e1) {
  int t = blockIdx.x * blockDim.x + threadIdx.x;   // over E*C
  if (t >= EN * CN) return;
  int e = t >> 6, c = t & 63;
  int s = eidx[e], d = eidx[EN + e];
  size_t p = (size_t)s * VN + d;
  e1[p * CN + c] = dual_attr[(size_t)e * CN + c];
  if (c == 0) adj0[p] = 1.0f;
}

__global__ void init_feat0(const float* __restrict__ adj, const float* __restrict__ x,
                           float* __restrict__ feat) {
  size_t t = (size_t)blockIdx.x * blockDim.x + threadIdx.x;  // over V*V*C
  if (t >= (size_t)VN * VN * CN) return;
  size_t p = t >> 6; int c = t & 63; int y = (int)(p % VN);
  feat[t] = adj[p] * x[y * CN + c];
}

__global__ void mask_update(float* __restrict__ nvn, const float* __restrict__ a0,
                            const float* __restrict__ a1, const float* __restrict__ a2,
                            int nprev, float* __restrict__ adjN,
                            float* __restrict__ nvp, float* __restrict__ nvsave) {
  int t = blockIdx.x * blockDim.x + threadIdx.x;   // over V*V
  if (t >= VN * VN) return;
  int i = t / VN, j = t % VN;
  float nv = nvn[t];
  bool m = nv > 0.0f;
  if (nprev > 0) m = m && !(a0[t] > 0.0f);
  if (nprev > 1) m = m && !(a1[t] > 0.0f);
  if (nprev > 2) m = m && !(a2[t] > 0.0f);
  m = m && (i != j);
  float an = m ? 1.0f : 0.0f;
  float nvm = nv * an;
  adjN[t] = an;
  nvn[t]  = nvm;
  nvp[t]  = nvm + an;
  if (nvsave) nvsave[t] = nvm;
}

// featN <- featN * adjN ;  featPres <- masked + adjN * x   (featPres may alias featN)
__global__ void feat_mask_add(float* __restrict__ featN, float* __restrict__ featPres,
                              const float* __restrict__ adjN, const float* __restrict__ x) {
  size_t t = (size_t)blockIdx.x * blockDim.x + threadIdx.x;
  if (t >= (size_t)VN * VN * CN) return;
  size_t p = t >> 6; int c = t & 63; int y = (int)(p % VN);
  float m = adjN[p];
  float f = featN[t] * m;
  featN[t] = f;
  featPres[t] = f + m * x[y * CN + c];
}

__global__ void normalize_eattr(const float* __restrict__ featSaved,
                                const float* __restrict__ numv,
                                const float* __restrict__ adjN,
                                float* __restrict__ eat) {
  size_t t = (size_t)blockIdx.x * blockDim.x + threadIdx.x;
  if (t >= (size_t)VN * VN * CN) return;
  size_t p = t >> 6;
  eat[t] = featSaved[t] / (numv[p] + 1e-7f) * adjN[p];
}

// ---------------- GAT pieces ----------------
__global__ void gat_logits(const float* __restrict__ s, const float* __restrict__ efull,
                           const float* __restrict__ adj, const float* __restrict__ avec,
                           float* __restrict__ logits) {
  size_t t = (size_t)blockIdx.x * blockDim.x + threadIdx.x;  // over V*V*H
  if (t >= (size_t)VN * VN * HN) return;
  int h = (int)(t & 15);
  size_t ij = t >> 4;
  int j = (int)(ij % VN), i = (int)(ij / VN);
  float acc = 0.0f;
#pragma unroll
  for (int d = 0; d < DN; ++d) {
    float z = s[i * CN + h * DN + d] + s[j * CN + h * DN + d] +
              efull[ij * CN + h * DN + d];
    z = z > 0.0f ? z : 0.2f * z;                 // leaky_relu(0.2)
    acc += avec[h * DN + d] * z;
  }
  bool m = (adj[ij] > 0.0f) || (i == j);          // adj + eye
  logits[ij * HN + h] = m ? acc : -1e9f;
}

// softmax over j (512) per (i,h): one wave each, 16 values per lane
__global__ __launch_bounds__(256) void gat_softmax(float* __restrict__ logits) {
  int wid = blockIdx.x * 8 + (threadIdx.x >> 5);  // over V*H = 8192
  int l = threadIdx.x & 31;
  int i = wid >> 4, h = wid & 15;
  float vals[16];
  float mx = -3.0e38f;
#pragma unroll
  for (int q = 0; q < 16; ++q) {
    int j = l + q * 32;
    vals[q] = logits[((size_t)i * VN + j) * HN + h];
    mx = fmaxf(mx, vals[q]);
  }
  mx = wave_max(mx);
  float sum = 0.0f;
#pragma unroll
  for (int q = 0; q < 16; ++q) { vals[q] = __expf(vals[q] - mx); sum += vals[q]; }
  sum = wave_sum(sum);
  float inv = 1.0f / sum;
#pragma unroll
  for (int q = 0; q < 16; ++q) {
    int j = l + q * 32;
    logits[((size_t)i * VN + j) * HN + h] = vals[q] * inv;
  }
}

__global__ void gat_aggregate(const float* __restrict__ alpha, const float* __restrict__ s,
                              float* __restrict__ xout) {
  int t = blockIdx.x * blockDim.x + threadIdx.x;  // over V*H*D = 32768
  if (t >= VN * HN * DN) return;
  int d = t & 3, h = (t >> 2) & 15, i = t >> 6;
  float acc = 0.0f;
  for (int j = 0; j < VN; ++j)
    acc += alpha[((size_t)i * VN + j) * HN + h] * s[j * CN + h * DN + d];
  xout[i * CN + h * DN + d] = acc > 0.0f ? acc : __expf(acc) - 1.0f;  // elu
}

// ---------------- fuse / edge-MLP / output ----------------
__global__ void concat3(const float* __restrict__ x1, const float* __restrict__ x2,
                        const float* __restrict__ x3, float* __restrict__ xcat) {
  int t = blockIdx.x * blockDim.x + threadIdx.x;  // over V*192
  if (t >= VN * 3 * CN) return;
  int i = t / (3 * CN), k = t % (3 * CN);
  float v = (k < CN) ? x1[i * CN + k]
          : (k < 2 * CN) ? x2[i * CN + k - CN] : x3[i * CN + k - 2 * CN];
  xcat[t] = v;
}

__global__ void add_bias(float* __restrict__ p, const float* __restrict__ b, int n) {
  int t = blockIdx.x * blockDim.x + threadIdx.x;
  if (t >= n) return;
  p[t] += b[t & 63];
}

// per-edge wave32 LayerNorm(128) + ReLU on concatenated endpoint feats
__global__ __launch_bounds__(256) void ln_relu_pairs(
    const float* __restrict__ xf, const int* __restrict__ eidx,
    const float* __restrict__ g, const float* __restrict__ b,
    float* __restrict__ out) {
  int e = blockIdx.x * 8 + (threadIdx.x >> 5);    // over E = 3072
  int l = threadIdx.x & 31;
  int sn = eidx[e], dn = eidx[EN + e];
  float p[4];
#pragma unroll
  for (int q = 0; q < 4; ++q) {
    int idx = l + q * 32;
    p[q] = (idx < CN) ? xf[sn * CN + idx] : xf[dn * CN + idx - CN];
  }
  float mu = wave_sum(p[0] + p[1] + p[2] + p[3]) * (1.0f / 128.0f);
  float var = 0.0f;
#pragma unroll
  for (int q = 0; q < 4; ++q) { float d = p[q] - mu; var += d * d; }
  var = wave_sum(var) * (1.0f / 128.0f);
  float inv = rsqrtf(var + 1e-5f);
#pragma unroll
  for (int q = 0; q < 4; ++q) {
    int idx = l + q * 32;
    float ln = (p[q] - mu) * inv * g[idx] + b[idx];
    out[(size_t)e * 128 + idx] = fmaxf(ln, 0.0f);
  }
}

__global__ void nea_out(const float* __restrict__ edge_attr, const float* __restrict__ delta,
                        const float* __restrict__ bm, float* __restrict__ nea) {
  int t = blockIdx.x * blockDim.x + threadIdx.x;  // over E*C
  if (t >= EN * CN) return;
  nea[t] = edge_attr[t] + delta[t] + bm[t & 63];
}

__global__ void gather_dual(const int* __restrict__ dualIdx, const float* __restrict__ nea,
                            float* __restrict__ fplan) {
  int t = blockIdx.x * blockDim.x + threadIdx.x;  // over EF*C
  if (t >= EFN * CN) return;
  int f = t >> 6, c = t & 63;
  int idx = dualIdx[f];
  fplan[t] = (idx >= EN) ? 1.0f : nea[(size_t)idx * CN + c];
}

// ---------------- host launcher ----------------
extern "C" void kernel_launch(void* const* d_in, const int* in_sizes, int n_in,
                              void* d_out, int out_size, void* d_ws, size_t ws_size,
                              hipStream_t stream) {
  (void)in_sizes; (void)n_in; (void)out_size; (void)ws_size;
  const float* x        = (const float*)d_in[0];
  const int*   eidx     = (const int*)d_in[1];
  const float* dualAttr = (const float*)d_in[2];
  const float* edgeAttr = (const float*)d_in[3];
  const int*   dualIdx  = (const int*)d_in[4];
  const float* W1 = (const float*)d_in[5],  *We1 = (const float*)d_in[6],  *a1 = (const float*)d_in[7];
  const float* W2 = (const float*)d_in[8],  *We2 = (const float*)d_in[9],  *a2 = (const float*)d_in[10];
  const float* W3 = (const float*)d_in[11], *We3 = (const float*)d_in[12], *a3 = (const float*)d_in[13];
  const float* Wf = (const float*)d_in[14], *bf = (const float*)d_in[15];
  const float* lng = (const float*)d_in[16], *lnb = (const float*)d_in[17];
  const float* Wm = (const float*)d_in[18], *bm = (const float*)d_in[19];

  const size_t VV = (size_t)VN * VN, VVC = VV * CN;
  float* ws   = (float*)d_ws;
  float* adj0 = ws;          float* adj1 = adj0 + VV;
  float* adj2 = adj1 + VV;   float* adj3 = adj2 + VV;
  float* nvp  = adj3 + VV;   float* nvn  = nvp + VV;
  float* nv1  = nvn + VV;
  float* bufA = nv1 + VV;
  float* bufB = bufA + VVC;
  float* bufC = bufB + VVC;
  float* eat  = bufC + VVC;
  float* efull= eat + VVC;
  float* logits = efull + VVC;                 // V*V*H
  float* sbuf = logits + VV * HN;
  float* x1b  = sbuf + VN * CN;
  float* x2b  = x1b + VN * CN;
  float* x3b  = x2b + VN * CN;
  float* xcat = x3b + VN * CN;                 // V*192
  float* lnbuf= xcat + (size_t)VN * 3 * CN;    // E*128
  float* delta= lnbuf + (size_t)EN * 128;      // E*64

  float* out   = (float*)d_out;
  float* fplan = out;                          // [EF, C]
  float* xf    = out + (size_t)EFN * CN;       // [V, H*D]
  float* nea   = xf + (size_t)VN * CN;         // [E, C]

  const dim3 B256(256);
  auto g1 = [](size_t n) { return dim3((unsigned)((n + 255) / 256)); };

  // ---- build dense adjacency + e1 ----
  hipMemsetAsync(adj0, 0, VV * sizeof(float), stream);
  hipMemsetAsync(eat, 0, VVC * sizeof(float), stream);   // e1 dense
  scatter_edges<<<g1((size_t)EN * CN), B256, 0, stream>>>(eidx, dualAttr, adj0, eat);
  hipMemcpyAsync(nvp, adj0, VV * sizeof(float), hipMemcpyDeviceToDevice, stream);
  init_feat0<<<g1(VVC), B256, 0, stream>>>(adj0, x, bufA);

  const dim3 pgGrid(VN / 16, VN / 16, CN / 8), pgBlk(256);
  const dim3 nvGrid(VN / 16, VN / 64), gemmBlk(128);

  // ---- hop 1 ----
  pathway_gemm<<<pgGrid, pgBlk, 0, stream>>>(bufA, adj0, bufB);
  wmma_gemm_rm<<<nvGrid, gemmBlk, 0, stream>>>(nvp, adj0, nvn, VN, VN, VN);
  mask_update<<<g1(VV), B256, 0, stream>>>(nvn, adj0, adj0, adj0, 1, adj1, nvp, nv1);
  feat_mask_add<<<g1(VVC), B256, 0, stream>>>(bufB, bufC, adj1, x);  // bufB = F1 kept
  // ---- hop 2 ----
  pathway_gemm<<<pgGrid, pgBlk, 0, stream>>>(bufC, adj1, bufA);
  wmma_gemm_rm<<<nvGrid, gemmBlk, 0, stream>>>(nvp, adj1, nvn, VN, VN, VN);
  mask_update<<<g1(VV), B256, 0, stream>>>(nvn, adj0, adj1, adj1, 2, adj2, nvp, nullptr);
  feat_mask_add<<<g1(VVC), B256, 0, stream>>>(bufA, bufA, adj2, x); // pres2 in place
  // ---- hop 3 ----
  pathway_gemm<<<pgGrid, pgBlk, 0, stream>>>(bufA, adj2, bufC);
  wmma_gemm_rm<<<nvGrid, gemmBlk, 0, stream>>>(nvp, adj2, nvn, VN, VN, VN);
  mask_update<<<g1(VV), B256, 0, stream>>>(nvn, adj0, adj1, adj2, 3, adj3, nvp, nullptr);
  feat_mask_add<<<g1(VVC), B256, 0, stream>>>(bufC, bufA, adj3, x); // bufC = F3, bufA dead

  const dim3 sGrid(CN / 16, VN / 64);
  const dim3 eGrid(CN / 16, (unsigned)(VV / 64));
  const size_t nLog = VV * HN;

  // ---- GAT 1 (1-hop, e1) ----
  wmma_gemm_rm<<<sGrid, gemmBlk, 0, stream>>>(x, W1, sbuf, VN, CN, CN);
  wmma_gemm_rm<<<eGrid, gemmBlk, 0, stream>>>(eat, We1, efull, (int)VV, CN, CN);
  gat_logits<<<g1(nLog), B256, 0, stream>>>(sbuf, efull, adj0, a1, logits);
  gat_softmax<<<dim3(VN * HN / 8), B256, 0, stream>>>(logits);
  gat_aggregate<<<g1(VN * HN * DN), B256, 0, stream>>>(logits, sbuf, x1b);
  // ---- GAT 2 (2-hop) ----
  normalize_eattr<<<g1(VVC), B256, 0, stream>>>(bufB, nv1, adj1, eat);
  wmma_gemm_rm<<<sGrid, gemmBlk, 0, stream>>>(x, W2, sbuf, VN, CN, CN);
  wmma_gemm_rm<<<eGrid, gemmBlk, 0, stream>>>(eat, We2, efull, (int)VV, CN, CN);
  gat_logits<<<g1(nLog), B256, 0, stream>>>(sbuf, efull, adj1, a2, logits);
  gat_softmax<<<dim3(VN * HN / 8), B256, 0, stream>>>(logits);
  gat_aggregate<<<g1(VN * HN * DN), B256, 0, stream>>>(logits, sbuf, x2b);
  // ---- GAT 3 (4-hop) ----
  normalize_eattr<<<g1(VVC), B256, 0, stream>>>(bufC, nvn, adj3, eat);
  wmma_gemm_rm<<<sGrid, gemmBlk, 0, stream>>>(x, W3, sbuf, VN, CN, CN);
  wmma_gemm_rm<<<eGrid, gemmBlk, 0, stream>>>(eat, We3, efull, (int)VV, CN, CN);
  gat_logits<<<g1(nLog), B256, 0, stream>>>(sbuf, efull, adj3, a3, logits);
  gat_softmax<<<dim3(VN * HN / 8), B256, 0, stream>>>(logits);
  gat_aggregate<<<g1(VN * HN * DN), B256, 0, stream>>>(logits, sbuf, x3b);

  // ---- fuse: xf = concat(x1,x2,x3) @ Wf + bf  (written to d_out slice) ----
  concat3<<<g1((size_t)VN * 3 * CN), B256, 0, stream>>>(x1b, x2b, x3b, xcat);
  wmma_gemm_rm<<<sGrid, gemmBlk, 0, stream>>>(xcat, Wf, xf, VN, CN, 3 * CN);
  add_bias<<<g1((size_t)VN * CN), B256, 0, stream>>>(xf, bf, VN * CN);

  // ---- edge MLP ----
  ln_relu_pairs<<<dim3(EN / 8), B256, 0, stream>>>(xf, eidx, lng, lnb, lnbuf);
  wmma_gemm_rm<<<dim3(CN / 16, EN / 64), gemmBlk, 0, stream>>>(lnbuf, Wm, delta, EN, CN, 2 * CN);
  nea_out<<<g1((size_t)EN * CN), B256, 0, stream>>>(edgeAttr, delta, bm, nea);

  // ---- padded gather to fplan_dual ----
  gather_dual<<<g1((size_t)EFN * CN), B256, 0, stream>>>(dualIdx, nea, fplan);
}